// SinkhornOT_6313601925278
// MI455X (gfx1250) — compile-verified
//
#include <hip/hip_runtime.h>
#include <math.h>

// CDNA5 / gfx1250, wave32. One block (8 waves) per batch item.
// GEMMs via V_WMMA_F32_16X16X4_F32 (exact f32 — Sinkhorn divides cost by
// eps=0.05, amplifying any C error x20, so low-precision WMMA is unsafe).
// All reduction phases (norms / Sinkhorn logsumexp / MESH) are parallelized
// across all 256 threads with a 4-quadrant decomposition so no wave idles.

typedef float v2f __attribute__((ext_vector_type(2)));
typedef float v8f __attribute__((ext_vector_type(8)));

#define DDIM 256
#define PSTR 260   // padded row stride for 64x256 LDS matrices (conflict-free frags)
#define CSTR 66    // padded row stride for 64x64 LDS matrices
#define EPS_INV 20.0f
#define SINK_ITERS 20
#define MESH_ITERS 3

static __device__ __forceinline__ v8f wmma_f32(v2f a, v2f b, v8f c) {
  // 8-arg f32 pattern: (neg_a, A, neg_b, B, c_mod, C, reuse_a, reuse_b)
  return __builtin_amdgcn_wmma_f32_16x16x4_f32(false, a, false, b, (short)0, c,
                                               false, false);
}

// Out[64x256] = act(Alds[64x256] @ W[256x256] + bias), A/Out in padded LDS.
// Wave w computes all 4 m-tiles for n-tiles {2w, 2w+1}: each W element is
// read from global (L2) exactly once per block.
static __device__ void gemm64(const float* __restrict__ Alds,
                              const float* __restrict__ W,
                              const float* __restrict__ bias,
                              float* __restrict__ Olds,
                              bool do_relu, int lane, int wave) {
  const int nidx = lane & 15;
  const int nhalf = lane >> 4;   // 0 or 1
  const int nt0 = wave * 2;

  v8f acc[4][2];
  for (int j = 0; j < 2; ++j) {
    const float bv = bias[(nt0 + j) * 16 + nidx];   // C layout: col = lane&15
    for (int mt = 0; mt < 4; ++mt)
      for (int r = 0; r < 8; ++r) acc[mt][j][r] = bv;
  }

  for (int kk = 0; kk < DDIM; kk += 4) {
    const int k0 = kk + 2 * nhalf;                  // A/B frag K split across half-waves
    v2f bf[2];
    for (int j = 0; j < 2; ++j) {
      const int n = (nt0 + j) * 16 + nidx;
      bf[j][0] = W[k0 * DDIM + n];                  // B[k][n]
      bf[j][1] = W[(k0 + 1) * DDIM + n];
    }
    for (int mt = 0; mt < 4; ++mt) {
      const float* ap = Alds + (mt * 16 + nidx) * PSTR + k0;  // A[m][k0..k0+1]
      v2f af;
      af[0] = ap[0];
      af[1] = ap[1];
      acc[mt][0] = wmma_f32(af, bf[0], acc[mt][0]);
      acc[mt][1] = wmma_f32(af, bf[1], acc[mt][1]);
    }
  }
  __syncthreads();  // all waves done reading Alds (Olds may alias Alds)
  for (int mt = 0; mt < 4; ++mt)
    for (int j = 0; j < 2; ++j)
      for (int r = 0; r < 8; ++r) {
        const int m = mt * 16 + nhalf * 8 + r;      // C/D layout
        const int n = (nt0 + j) * 16 + nidx;
        float v = acc[mt][j][r];
        if (do_relu) v = fmaxf(v, 0.0f);
        Olds[m * PSTR + n] = v;
      }
  __syncthreads();
}

__global__ __launch_bounds__(256) void sinkhorn_ot_fused(
    const float* __restrict__ slots_q, const float* __restrict__ slots_r,
    const float* __restrict__ mask_q, const float* __restrict__ mask_r,
    const float* __restrict__ W1, const float* __restrict__ b1,
    const float* __restrict__ W2, const float* __restrict__ b2,
    float* __restrict__ out, int B) {
  __shared__ float s_qp[64 * PSTR];    // q projection
  __shared__ float s_rp[64 * PSTR];    // r projection
  __shared__ float s_work[64 * PSTR];  // X/H scratch; later aliased as C and T
  __shared__ float s_la[64], s_lb[64], s_lmq[64], s_lmr[64];
  __shared__ float s_q2[64], s_r2[64], s_rs[64];
  __shared__ float s_red_mx[256], s_red_sm[256];

  const int b = blockIdx.x;
  const int tid = threadIdx.x;
  const int lane = tid & 31;
  const int wave = tid >> 5;
  const int idx = tid & 63;   // row (or column) owned in reduction phases
  const int qq = tid >> 6;    // quadrant: which 16-element slice of the 64

  float* const sC = s_work;                 // 64 x CSTR
  float* const sT = s_work + 64 * CSTR;     // 64 x CSTR

  float* out_sim = out;                                        // [B]
  float* out_T = out + (size_t)B;                              // [B,64,64]
  float* out_C = out + (size_t)B + (size_t)B * 4096;           // [B,64,64]
  float* out_cost = out + (size_t)B + (size_t)2 * B * 4096;    // [B]

  // ---------------- projections: x -> relu(xW1+b1)W2+b2 ----------------
  for (int which = 0; which < 2; ++which) {
    const float* src = which ? slots_r : slots_q;
    float* dst = which ? s_rp : s_qp;
    for (int i = tid; i < 64 * DDIM; i += 256)
      s_work[(i >> 8) * PSTR + (i & 255)] = src[(size_t)b * 64 * DDIM + i];
    __syncthreads();
    gemm64(s_work, W1, b1, s_work, true, lane, wave);   // H = relu(X W1 + b1)
    gemm64(s_work, W2, b2, dst, false, lane, wave);     // P = H W2 + b2
  }

  // ------------- row norms (all 256 threads, b128 sweeps) + log-masks ----
  {
    const float4* qrow = (const float4*)&s_qp[idx * PSTR + qq * 64];
    const float4* rrow = (const float4*)&s_rp[idx * PSTR + qq * 64];
    float sq = 0.f, sr = 0.f;
    for (int c = 0; c < 16; ++c) {
      const float4 a = qrow[c];
      const float4 d = rrow[c];
      sq += a.x * a.x + a.y * a.y + a.z * a.z + a.w * a.w;
      sr += d.x * d.x + d.y * d.y + d.z * d.z + d.w * d.w;
    }
    s_red_mx[tid] = sq;
    s_red_sm[tid] = sr;
    __syncthreads();
    if (tid < 64) {
      s_q2[tid] = s_red_mx[tid] + s_red_mx[tid + 64] + s_red_mx[tid + 128] +
                  s_red_mx[tid + 192];
      s_la[tid] = 0.f;
      s_lb[tid] = 0.f;
    } else if (tid < 128) {
      const int r = tid - 64;
      s_r2[r] = s_red_sm[r] + s_red_sm[r + 64] + s_red_sm[r + 128] +
                s_red_sm[r + 192];
    } else if (tid < 192) {
      const int k = tid - 128;
      s_lmq[k] = logf(fmaxf(mask_q[(size_t)b * 64 + k], 1e-8f));
    } else {
      const int m = tid - 192;
      s_lmr[m] = logf(fmaxf(mask_r[(size_t)b * 64 + m], 1e-8f));
    }
    __syncthreads();
  }

  // ---------------- QR gram + C = sqrt(clip(q2 + r2 - 2 QR)) -------------
  {
    const int nidx = lane & 15, nhalf = lane >> 4;
    const int mt = wave & 3;
    const int nt0 = (wave >> 2) * 2;
    v8f acc[2];
    for (int j = 0; j < 2; ++j)
      for (int r = 0; r < 8; ++r) acc[j][r] = 0.f;
    for (int kk = 0; kk < DDIM; kk += 4) {
      const int k0 = kk + 2 * nhalf;
      const float* ap = s_qp + (mt * 16 + nidx) * PSTR + k0;
      v2f af;
      af[0] = ap[0];
      af[1] = ap[1];
      for (int j = 0; j < 2; ++j) {
        // B[k][n] = r_proj[n][k] (transpose read straight from padded LDS)
        const float* bp = s_rp + ((nt0 + j) * 16 + nidx) * PSTR + k0;
        v2f bf;
        bf[0] = bp[0];
        bf[1] = bp[1];
        acc[j] = wmma_f32(af, bf, acc[j]);
      }
    }
    for (int j = 0; j < 2; ++j)
      for (int r = 0; r < 8; ++r) {
        const int m = mt * 16 + nhalf * 8 + r;
        const int n = (nt0 + j) * 16 + nidx;
        const float d2 = s_q2[m] + s_r2[n] - 2.0f * acc[j][r];
        const float cv = sqrtf(fmaxf(d2, 0.0f));
        sC[m * CSTR + n] = cv;
        out_C[(size_t)b * 4096 + m * 64 + n] = cv;
      }
  }
  __syncthreads();

  // ------------- log-domain Sinkhorn (20 iters, all threads active) ------
  for (int it = 0; it < SINK_ITERS; ++it) {
    // la[k] = -logsumexp_m(logK[k,m] + lb[m]); thread owns (k=idx, 16 cols)
    {
      const int k = idx;
      const float base = s_lmq[k];
      float vbuf[16];
      float pm = -3.4e38f;
#pragma unroll
      for (int i = 0; i < 16; ++i) {
        const int m = qq * 16 + i;
        const float v =
            fmaf(-EPS_INV, sC[k * CSTR + m], base + s_lmr[m] + s_lb[m]);
        vbuf[i] = v;
        pm = fmaxf(pm, v);
      }
      s_red_mx[tid] = pm;
      __syncthreads();
      const float mx = fmaxf(fmaxf(s_red_mx[k], s_red_mx[k + 64]),
                             fmaxf(s_red_mx[k + 128], s_red_mx[k + 192]));
      float ps = 0.f;
#pragma unroll
      for (int i = 0; i < 16; ++i) ps += expf(vbuf[i] - mx);
      s_red_sm[tid] = ps;
      __syncthreads();
      if (qq == 0) {
        const float s = s_red_sm[k] + s_red_sm[k + 64] + s_red_sm[k + 128] +
                        s_red_sm[k + 192];
        s_la[k] = -(mx + logf(s));
      }
      __syncthreads();
    }
    // lb[m] = -logsumexp_k(logK[k,m] + la[k]); thread owns (m=idx, 16 rows)
    {
      const int m = idx;
      const float base = s_lmr[m];
      float vbuf[16];
      float pm = -3.4e38f;
#pragma unroll
      for (int i = 0; i < 16; ++i) {
        const int k = qq * 16 + i;
        const float v =
            fmaf(-EPS_INV, sC[k * CSTR + m], base + s_lmq[k] + s_la[k]);
        vbuf[i] = v;
        pm = fmaxf(pm, v);
      }
      s_red_mx[tid] = pm;
      __syncthreads();
      const float mx = fmaxf(fmaxf(s_red_mx[m], s_red_mx[m + 64]),
                             fmaxf(s_red_mx[m + 128], s_red_mx[m + 192]));
      float ps = 0.f;
#pragma unroll
      for (int i = 0; i < 16; ++i) ps += expf(vbuf[i] - mx);
      s_red_sm[tid] = ps;
      __syncthreads();
      if (qq == 0) {
        const float s = s_red_sm[m] + s_red_sm[m + 64] + s_red_sm[m + 128] +
                        s_red_sm[m + 192];
        s_lb[m] = -(mx + logf(s));
      }
      __syncthreads();
    }
  }

  // T = exp(logK + la + lb)
  for (int i = tid; i < 4096; i += 256) {
    const int k = i >> 6, m = i & 63;
    const float v = fmaf(-EPS_INV, sC[k * CSTR + m],
                         s_lmq[k] + s_lmr[m] + s_la[k] + s_lb[m]);
    sT[k * CSTR + m] = expf(v);
  }
  __syncthreads();

  // ------------- MESH sharpening (3 iters, all threads active) -----------
  for (int it = 0; it < MESH_ITERS; ++it) {
    // square + row sums: thread owns (k=idx, 16 cols)
    {
      const int k = idx;
      float ps = 0.f;
#pragma unroll
      for (int i = 0; i < 16; ++i) {
        const int m = qq * 16 + i;
        float t = sT[k * CSTR + m];
        t *= t;
        sT[k * CSTR + m] = t;
        ps += t;
      }
      s_red_sm[tid] = ps;
      __syncthreads();
      if (qq == 0)
        s_rs[k] = s_red_sm[k] + s_red_sm[k + 64] + s_red_sm[k + 128] +
                  s_red_sm[k + 192];
      __syncthreads();
    }
    // row-normalize (kept in regs) + col sums, then col-normalize
    {
      const int m = idx;
      float tbuf[16];
      float cs = 0.f;
#pragma unroll
      for (int i = 0; i < 16; ++i) {
        const int k = qq * 16 + i;
        const float t = sT[k * CSTR + m] / (s_rs[k] + 1e-8f);
        tbuf[i] = t;
        cs += t;
      }
      s_red_sm[tid] = cs;
      __syncthreads();
      const float den = (s_red_sm[m] + s_red_sm[m + 64] + s_red_sm[m + 128] +
                         s_red_sm[m + 192]) +
                        1e-8f;
#pragma unroll
      for (int i = 0; i < 16; ++i)
        sT[(qq * 16 + i) * CSTR + m] = tbuf[i] / den;
      __syncthreads();
    }
  }

  // ---------------- outputs: T_hard, transport cost, similarity ----------
  float part = 0.f;
  for (int i = tid; i < 4096; i += 256) {
    const int k = i >> 6, m = i & 63;
    const float t = sT[k * CSTR + m];
    part += t * sC[k * CSTR + m];
    out_T[(size_t)b * 4096 + i] = t;
  }
  s_red_mx[tid] = part;
  __syncthreads();
  if (tid < 64)
    s_red_mx[tid] = s_red_mx[tid] + s_red_mx[tid + 64] + s_red_mx[tid + 128] +
                    s_red_mx[tid + 192];
  __syncthreads();
  if (tid == 0) {
    float s = 0.f;
    for (int i = 0; i < 64; ++i) s += s_red_mx[i];
    out_cost[b] = s;
    out_sim[b] = 1.0f / (1.0f + expf(s));  // sigmoid(-cost)
  }
}

extern "C" void kernel_launch(void* const* d_in, const int* in_sizes, int n_in,
                              void* d_out, int out_size, void* d_ws,
                              size_t ws_size, hipStream_t stream) {
  const float* slots_q = (const float*)d_in[0];
  const float* slots_r = (const float*)d_in[1];
  const float* mask_q = (const float*)d_in[2];
  const float* mask_r = (const float*)d_in[3];
  const float* W1 = (const float*)d_in[4];
  const float* b1 = (const float*)d_in[5];
  const float* W2 = (const float*)d_in[6];
  const float* b2 = (const float*)d_in[7];
  float* out = (float*)d_out;

  const int B = in_sizes[0] / (64 * DDIM);  // 2048
  (void)d_ws;
  (void)ws_size;
  (void)n_in;
  (void)out_size;

  hipLaunchKernelGGL(sinkhorn_ot_fused, dim3(B), dim3(256), 0, stream, slots_q,
                     slots_r, mask_q, mask_r, W1, b1, W2, b2, out, B);
}